// WindowLayer_54966991454546
// MI455X (gfx1250) — compile-verified
//
#include <hip/hip_runtime.h>

typedef __attribute__((ext_vector_type(2))) float v2f;
typedef __attribute__((ext_vector_type(8))) float v8f;

#define B_SZ  4096
#define H_SZ  1024
#define K_MIX 10
#define L_SZ  64
#define V_SZ  80
#define NC    30          // 3*K columns of the linear layer
#define WAVES 8           // 256 threads / wave32

__global__ __launch_bounds__(256)
void window_layer_kernel(const float* __restrict__ x,
                         const float* __restrict__ chars,
                         const float* __restrict__ W,
                         const float* __restrict__ bias,
                         float* __restrict__ out)
{
    // LDS staging: per-wave partial C tiles, exp'd alpha/beta/kappa, phi
    __shared__ float s_part[WAVES][16][32];   // 16 KB
    __shared__ float s_abk[16][32];           //  2 KB
    __shared__ float s_phi[16][L_SZ];         //  4 KB

    const int tid  = threadIdx.x;
    const int lane = tid & 31;
    const int wave = tid >> 5;
    const int half = lane >> 4;     // 0: lanes 0-15, 1: lanes 16-31
    const int lcol = lane & 15;
    const int row0 = blockIdx.x * 16;   // 16 batch rows per block

    // ---- Phase 0: prefetch this block's chars tile (320 KB) into L2 --------
    // MI455X L2 is 192 MB: the whole 84 MB chars tensor is L2-resident, so
    // pulling it in now overlaps the dominant HBM stream with the GEMM/phi
    // phases. Emits global_prefetch_b8; no counters, fire-and-forget.
    {
        const char* cp = (const char*)(chars + (size_t)row0 * L_SZ * V_SZ);
        const int   tile_bytes = 16 * L_SZ * V_SZ * 4;          // 327680
        for (int off = tid * 128; off < tile_bytes; off += 256 * 128)
            __builtin_prefetch(cp + off, 0, 3);
    }

    // ---------------- Phase 1: C = x @ W  via V_WMMA_F32_16X16X4_F32 ----------
    // Wave-level K-split: each wave accumulates a 128-deep chunk of H.
    v8f c0 = {};   // output cols 0..15
    v8f c1 = {};   // output cols 16..31 (padding cols carry garbage, never read)
    const int kbeg = wave * (H_SZ / WAVES);
    const int kend = kbeg + (H_SZ / WAVES);
    const float* xrow = x + (size_t)(row0 + lcol) * H_SZ;   // A: M = lcol
    // Branchless padding: clamp col 30/31 -> 29. Cols >= NC accumulate junk
    // that phases 2-4 never consume; keeps the loop free of exec-mask branches.
    const int n1 = (16 + lcol < NC) ? (16 + lcol) : (NC - 1);

    for (int k = kbeg; k < kend; k += 4) {
        const int krow = k + 2 * half;  // ISA 32-bit A layout: lanes 0-15 K=k,k+1 ; 16-31 K=k+2,k+3
        v2f a;
        a.x = xrow[krow + 0];
        a.y = xrow[krow + 1];

        v2f b0, b1;                     // B row-striped to match A's K grouping
        b0.x = W[(size_t)(krow + 0) * NC + lcol];
        b0.y = W[(size_t)(krow + 1) * NC + lcol];
        b1.x = W[(size_t)(krow + 0) * NC + n1];
        b1.y = W[(size_t)(krow + 1) * NC + n1];

        c0 = __builtin_amdgcn_wmma_f32_16x16x4_f32(false, a, false, b0,
                                                   (short)0, c0, false, false);
        c1 = __builtin_amdgcn_wmma_f32_16x16x4_f32(false, a, false, b1,
                                                   (short)0, c1, false, false);
    }

    // C/D layout: VGPR r, lanes 0-15 -> (M=r, N=lane); lanes 16-31 -> (M=r+8, N=lane-16)
    #pragma unroll
    for (int r = 0; r < 8; ++r) {
        const int rr = r + 8 * half;
        s_part[wave][rr][lcol]      = c0[r];
        s_part[wave][rr][lcol + 16] = c1[r];
    }
    __syncthreads();

    // ------------- Phase 2: deterministic cross-wave reduce + bias + exp ------
    // alpha = exp(raw), beta = exp(raw), kappa = exp(raw): exp applied here once.
    for (int i = tid; i < 16 * 32; i += 256) {
        const int row = i >> 5;
        const int col = i & 31;
        float s = (col < NC) ? bias[col] : 0.0f;
        #pragma unroll
        for (int w = 0; w < WAVES; ++w) s += s_part[w][row][col];
        s_abk[row][col] = __expf(s);
    }
    __syncthreads();

    // ------------- Phase 3: phi[row][l] = sum_k a_k * exp(-b_k*(k_k - l)^2) ---
    for (int i = tid; i < 16 * L_SZ; i += 256) {
        const int   row = i >> 6;
        const float lu  = (float)(i & 63);
        float acc = 0.0f;
        #pragma unroll
        for (int k = 0; k < K_MIX; ++k) {
            const float al = s_abk[row][k];
            const float be = s_abk[row][K_MIX + k];
            const float ka = s_abk[row][2 * K_MIX + k];
            const float d  = ka - lu;
            acc += al * __expf(-be * d * d);
        }
        s_phi[row][i & 63] = acc;
    }
    __syncthreads();

    // ------------- Phase 4: window[b][v] = sum_l phi[b][l] * chars[b][l][v] ---
    // float4 over v: full b128 loads/stores on the dominant 84 MB stream.
    const int NT4 = 16 * (V_SZ / 4);                 // 320 vec4 outputs per block
    for (int i = tid; i < NT4; i += 256) {
        const int row = i / (V_SZ / 4);
        const int v4  = (i % (V_SZ / 4)) * 4;
        const float4* cb =
            (const float4*)(chars + ((size_t)(row0 + row) * L_SZ) * V_SZ + v4);
        float4 acc = make_float4(0.f, 0.f, 0.f, 0.f);
        #pragma unroll 8
        for (int l = 0; l < L_SZ; ++l) {
            const float  p = s_phi[row][l];
            const float4 c = cb[(size_t)l * (V_SZ / 4)];
            acc.x += p * c.x;
            acc.y += p * c.y;
            acc.z += p * c.z;
            acc.w += p * c.w;
        }
        *(float4*)(out + (size_t)(row0 + row) * V_SZ + v4) = acc;
    }
}

extern "C" void kernel_launch(void* const* d_in, const int* in_sizes, int n_in,
                              void* d_out, int out_size, void* d_ws, size_t ws_size,
                              hipStream_t stream) {
    const float* x     = (const float*)d_in[0];
    const float* chars = (const float*)d_in[1];
    const float* W     = (const float*)d_in[2];
    const float* b     = (const float*)d_in[3];
    float* out = (float*)d_out;

    dim3 grid(B_SZ / 16);   // 256 workgroups, 16 batch rows each
    dim3 block(256);        // 8 wave32
    hipLaunchKernelGGL(window_layer_kernel, grid, block, 0, stream,
                       x, chars, W, b, out);
}